// GATNet_8504035246169
// MI455X (gfx1250) — compile-verified
//
#include <hip/hip_runtime.h>
#include <math.h>

// ---------------------------------------------------------------------------
// GAT (2 layers, edge features, softmax attention, max aggregation) for
// MI455X / gfx1250.  fp32 end-to-end; dense GEMMs use V_WMMA_F32_16X16X4_F32.
// Segment reductions use L2-resident native float atomics (acc = 12.8MB per
// layer, trivially resident in the 192MB L2).
// ---------------------------------------------------------------------------

typedef __attribute__((ext_vector_type(2))) float v2f;
typedef __attribute__((ext_vector_type(8))) float v8f;

#define DOUT 64
#define DE   16

// ---- native float atomic max / add (agent scope, relaxed) ------------------
// Targets GLOBAL_ATOMIC_MAX_NUM_F32 / GLOBAL_ATOMIC_ADD_F32 on gfx1250.
__device__ inline void atomicMaxF32(float* addr, float val) {
  __hip_atomic_fetch_max(addr, val, __ATOMIC_RELAXED, __HIP_MEMORY_SCOPE_AGENT);
}
__device__ inline void atomicAddF32(float* addr, float val) {
  __hip_atomic_fetch_add(addr, val, __ATOMIC_RELAXED, __HIP_MEMORY_SCOPE_AGENT);
}

// ---------------------------------------------------------------------------
// H[n,64] = X[n,64] @ W[64,64]   (row-major), exact fp32 via WMMA 16x16x4.
// block = 256 threads = 8 waves; wave w handles rows [blk*128 + w*16, +16).
// ---------------------------------------------------------------------------
__global__ void gemm64_wmma(const float* __restrict__ X,
                            const float* __restrict__ W,
                            float* __restrict__ H, int n) {
  __shared__ float Ws[64 * 64];
  for (int i = threadIdx.x; i < 64 * 64; i += blockDim.x) Ws[i] = W[i];
  __syncthreads();

  const int lane = threadIdx.x & 31;
  const int wave = threadIdx.x >> 5;
  const int hf   = lane >> 4;   // 0 = lanes 0-15, 1 = lanes 16-31
  const int l16  = lane & 15;

  const int rowBase = (blockIdx.x * 8 + wave) * 16;   // wave-uniform
  if (rowBase >= n) return;                           // uniform exit, EXEC stays full

  const int row  = rowBase + l16;
  const int rowc = (row < n) ? row : (n - 1);         // clamp so EXEC==all-1s at WMMA
  const float* xrow = X + (size_t)rowc * DOUT;

  for (int ct = 0; ct < 4; ++ct) {                    // 16-col output tiles
    v8f acc = {};
#pragma unroll
    for (int ks = 0; ks < 16; ++ks) {                 // K = 64 in steps of 4
      const int k0 = ks * 4 + 2 * hf;
      // A 16x4 fp32: v0 = {K0 | K2}, v1 = {K1 | K3} split by lane half
      v2f a = *(const v2f*)(xrow + k0);
      // B 4x16 fp32: v0 row K0|K2, v1 row K1|K3; col = l16
      v2f b;
      b.x = Ws[(k0 + 0) * 64 + ct * 16 + l16];
      b.y = Ws[(k0 + 1) * 64 + ct * 16 + l16];
      acc = __builtin_amdgcn_wmma_f32_16x16x4_f32(
          /*neg_a=*/false, a, /*neg_b=*/false, b,
          /*c_mod=*/(short)0, acc, /*reuse_a=*/false, /*reuse_b=*/false);
    }
    // C layout: VGPR r -> M = r (lanes 0-15) / M = r+8 (lanes 16-31); N = l16
#pragma unroll
    for (int r = 0; r < 8; ++r) {
      const int ro = rowBase + r + 8 * hf;
      if (ro < n) H[(size_t)ro * DOUT + ct * 16 + l16] = acc[r];
    }
  }
}

// ---- wae[k] = sum_c We[k][c] * a_e[c]  (folds e@a_e into a 16-dot per edge) ----
__global__ void compute_wae(const float* __restrict__ We,
                            const float* __restrict__ a_e,
                            float* __restrict__ wae) {
  const int k = threadIdx.x;
  if (k >= DE) return;
  float s = 0.f;
#pragma unroll
  for (int c = 0; c < DOUT; ++c) s += We[k * DOUT + c] * a_e[c];
  wae[k] = s;
}

// ---- hs[i] = h[i]·a_s, hd[i] = h[i]·a_d ----
__global__ void node_dots(const float* __restrict__ h,
                          const float* __restrict__ a_s,
                          const float* __restrict__ a_d,
                          float* __restrict__ hs, float* __restrict__ hd, int n) {
  const int i = blockIdx.x * blockDim.x + threadIdx.x;
  if (i >= n) return;
  const float* r = h + (size_t)i * DOUT;
  float s = 0.f, d = 0.f;
#pragma unroll
  for (int c = 0; c < DOUT; ++c) { const float v = r[c]; s += v * a_s[c]; d += v * a_d[c]; }
  hs[i] = s; hd[i] = d;
}

__global__ void fill_f32(float* __restrict__ p, long long n, float v) {
  const long long i = (long long)blockIdx.x * blockDim.x + threadIdx.x;
  if (i < n) p[i] = v;
}

// ---- pass A: per-edge attention logit + segment max over dst ----
__global__ void edge_logits(const int* __restrict__ src, const int* __restrict__ dst,
                            const float* __restrict__ attr, const float* __restrict__ wae,
                            const float* __restrict__ hs, const float* __restrict__ hd,
                            float* __restrict__ logits, float* __restrict__ m, int E) {
  const int e = blockIdx.x * blockDim.x + threadIdx.x;
  if (e >= E) return;
  const float4* ar = (const float4*)(attr + (size_t)e * DE);
  const float4* wr = (const float4*)wae;
  float el = 0.f;
#pragma unroll
  for (int q = 0; q < 4; ++q) {
    const float4 a = ar[q], w = wr[q];
    el += a.x * w.x + a.y * w.y + a.z * w.z + a.w * w.w;
  }
  const int s = src[e], d = dst[e];
  float lg = hs[s] + hd[d] + el;
  lg = (lg > 0.f) ? lg : 0.2f * lg;          // leaky_relu, ATT_SLOPE
  logits[e] = lg;
  atomicMaxF32(m + d, lg);
}

// ---- pass B: p = exp(logit - m[dst]); segment sum over dst ----
__global__ void edge_exp(const int* __restrict__ dst, const float* __restrict__ logits,
                         const float* __restrict__ m, float* __restrict__ p,
                         float* __restrict__ ssum, int E) {
  const int e = blockIdx.x * blockDim.x + threadIdx.x;
  if (e >= E) return;
  const int d = dst[e];
  const float pe = __expf(logits[e] - m[d]);
  p[e] = pe;
  atomicAddF32(ssum + d, pe);
}

// ---- rinv[i] = 1 / max(ssum[i], 1e-16): 50k divides instead of 102M ----
__global__ void node_recip(const float* __restrict__ ssum, float* __restrict__ rinv, int n) {
  const int i = blockIdx.x * blockDim.x + threadIdx.x;
  if (i >= n) return;
  rinv[i] = 1.0f / fmaxf(ssum[i], 1e-16f);
}

// ---- pass C: msg = (h[src]+e)*att, segment MAX into acc.  thread = (edge,col).
//      e_row recomputed on the fly (16 FMAs) from attr row + We staged in LDS. ----
__global__ void edge_message(const int* __restrict__ src, const int* __restrict__ dst,
                             const float* __restrict__ attr, const float* __restrict__ We,
                             const float* __restrict__ h, const float* __restrict__ p,
                             const float* __restrict__ rinv, float* __restrict__ acc,
                             long long total) {
  __shared__ float Wes[DE * DOUT];
  for (int i = threadIdx.x; i < DE * DOUT; i += blockDim.x) Wes[i] = We[i];
  __syncthreads();
  const long long idx = (long long)blockIdx.x * blockDim.x + threadIdx.x;
  if (idx >= total) return;
  const int e = (int)(idx >> 6);
  const int c = (int)(idx & 63);
  const int d = dst[e];
  const float att = p[e] * rinv[d];
  const float4* ar = (const float4*)(attr + (size_t)e * DE);
  float er = 0.f;
#pragma unroll
  for (int q = 0; q < 4; ++q) {
    const float4 a = ar[q];
    er += a.x * Wes[(4 * q + 0) * DOUT + c];
    er += a.y * Wes[(4 * q + 1) * DOUT + c];
    er += a.z * Wes[(4 * q + 2) * DOUT + c];
    er += a.w * Wes[(4 * q + 3) * DOUT + c];
  }
  const int s = src[e];
  const float msg = (h[(size_t)s * DOUT + c] + er) * att;
  atomicMaxF32(acc + (size_t)d * DOUT + c, msg);
}

// ---- pass D: zero empty segments (-inf), add bias, leaky_relu(0.01) ----
__global__ void node_finalize(const float* __restrict__ acc, const float* __restrict__ bias,
                              float* __restrict__ out, int n) {
  const int idx = blockIdx.x * blockDim.x + threadIdx.x;
  if (idx >= n * DOUT) return;
  const int c = idx & 63;
  float v = acc[idx];
  if (!__builtin_isfinite(v)) v = 0.f;
  v += bias[c];
  out[idx] = (v > 0.f) ? v : 0.01f * v;
}

// ---------------------------------------------------------------------------
static void run_layer(const float* Xin, const int* src, const int* dst,
                      const float* attr,
                      const float* W, const float* We, const float* a_s,
                      const float* a_d, const float* a_e, const float* bias,
                      float* Yout,
                      float* h, float* hs, float* hd, float* m, float* ssum,
                      float* wae, float* logits, float* pbuf, float* rinv,
                      float* acc, int N, int E, hipStream_t stream) {
  const long long NF = (long long)N * DOUT;
  const long long EC = (long long)E * DOUT;

  compute_wae<<<1, 32, 0, stream>>>(We, a_e, wae);
  gemm64_wmma<<<dim3((N + 127) / 128), 256, 0, stream>>>(Xin, W, h, N);
  node_dots<<<dim3((N + 255) / 256), 256, 0, stream>>>(h, a_s, a_d, hs, hd, N);

  fill_f32<<<dim3((unsigned)((N + 255) / 256)), 256, 0, stream>>>(m, N, -INFINITY);
  fill_f32<<<dim3((unsigned)((N + 255) / 256)), 256, 0, stream>>>(ssum, N, 0.f);
  fill_f32<<<dim3((unsigned)((NF + 255) / 256)), 256, 0, stream>>>(acc, NF, -INFINITY);

  edge_logits<<<dim3((E + 255) / 256), 256, 0, stream>>>(src, dst, attr, wae, hs, hd,
                                                         logits, m, E);
  edge_exp<<<dim3((E + 255) / 256), 256, 0, stream>>>(dst, logits, m, pbuf, ssum, E);
  node_recip<<<dim3((N + 255) / 256), 256, 0, stream>>>(ssum, rinv, N);
  edge_message<<<dim3((unsigned)((EC + 255) / 256)), 256, 0, stream>>>(
      src, dst, attr, We, h, pbuf, rinv, acc, EC);
  node_finalize<<<dim3((unsigned)((NF + 255) / 256)), 256, 0, stream>>>(acc, bias, Yout, N);
}

extern "C" void kernel_launch(void* const* d_in, const int* in_sizes, int n_in,
                              void* d_out, int out_size, void* d_ws, size_t ws_size,
                              hipStream_t stream) {
  const float* X    = (const float*)d_in[0];
  const int*   ei   = (const int*)d_in[1];
  const float* attr = (const float*)d_in[2];
  const float* W1   = (const float*)d_in[3];
  const float* We1  = (const float*)d_in[4];
  const float* as1  = (const float*)d_in[5];
  const float* ad1  = (const float*)d_in[6];
  const float* ae1  = (const float*)d_in[7];
  const float* b1   = (const float*)d_in[8];
  const float* W2   = (const float*)d_in[9];
  const float* We2  = (const float*)d_in[10];
  const float* as2  = (const float*)d_in[11];
  const float* ad2  = (const float*)d_in[12];
  const float* ae2  = (const float*)d_in[13];
  const float* b2   = (const float*)d_in[14];

  const int N = in_sizes[0] / DOUT;
  const int E = in_sizes[1] / 2;
  const int* src = ei;
  const int* dst = ei + E;

  // workspace layout (floats): ~52 MB total
  float* ws     = (float*)d_ws;
  float* h      = ws;                        // N*64
  float* c1     = h  + (size_t)N * DOUT;     // N*64
  float* acc    = c1 + (size_t)N * DOUT;     // N*64
  float* hs     = acc + (size_t)N * DOUT;    // N
  float* hd     = hs + N;                    // N
  float* m      = hd + N;                    // N
  float* ssum   = m  + N;                    // N
  float* rinv   = ssum + N;                  // N
  float* wae    = rinv + N;                  // 16
  float* logits = wae + 16;                  // E
  float* pbuf   = logits + E;                // E

  run_layer(X,  src, dst, attr, W1, We1, as1, ad1, ae1, b1, c1,
            h, hs, hd, m, ssum, wae, logits, pbuf, rinv, acc, N, E, stream);
  run_layer(c1, src, dst, attr, W2, We2, as2, ad2, ae2, b2, (float*)d_out,
            h, hs, hd, m, ssum, wae, logits, pbuf, rinv, acc, N, E, stream);
}